// BioSelfAttention_65506841198758
// MI455X (gfx1250) — compile-verified
//
#include <hip/hip_runtime.h>

typedef float v2f __attribute__((ext_vector_type(2)));
typedef float v8f __attribute__((ext_vector_type(8)));

#define LIF_STEPS 100
#define LIF_ALPHA 0.05f          // dt/tau = 0.001/0.02
#define WTA_INH  (-0.9f)
#define WTA_EXC  (1.1f)
#define WTA_STEPS 20

__device__ __forceinline__ float lif_rate(float J) {
    float v = 0.0f, cnt = 0.0f;
    for (int s = 0; s < LIF_STEPS; ++s) {
        v += LIF_ALPHA * (J - v);           // leaky integration
        if (v >= 1.0f) { cnt += 1.0f; v = 0.0f; }   // spike + reset
    }
    return cnt * (1.0f / LIF_STEPS);
}

// ---- Kernel A: per-token <Q_i,K_i> over D=64, then LIF -> rates (B*H*T = 4096)
__global__ void qk_lif_kernel(const float* __restrict__ Q,
                              const float* __restrict__ K,
                              float* __restrict__ rates, int n) {
    int idx = blockIdx.x * blockDim.x + threadIdx.x;
    if (idx >= n) return;
    const float4* q = reinterpret_cast<const float4*>(Q + (size_t)idx * 64);
    const float4* k = reinterpret_cast<const float4*>(K + (size_t)idx * 64);
    float dot = 0.0f;
#pragma unroll
    for (int i = 0; i < 16; ++i) {
        float4 a = q[i], b = k[i];
        dot += a.x * b.x + a.y * b.y + a.z * b.z + a.w * b.w;
    }
    rates[idx] = lif_rate(dot);
}

// ---- Kernel B: 3D WTA, 20 iters of clamp(x + x@W^T) with x:(32,128), W:(128,128).
// Real fp32 matmul via V_WMMA_F32_16X16X4_F32. W synthesized in registers.
// Old x is passed as the WMMA C accumulator so D = x + x@W^T directly.
__global__ __launch_bounds__(256) void wta3d_wmma_kernel(float* __restrict__ rates) {
    __shared__ float xs[32 * 128];          // 16 KB of the 320 KB WGP LDS
    const int tid  = threadIdx.x;
    const int wave = tid >> 5;              // 8 waves; wave == N-tile (N=128 -> 8 tiles)
    const int lane = tid & 31;
    const int lm   = lane & 15;
    const int half = lane >> 4;

    for (int i = tid; i < 32 * 128; i += 256) xs[i] = rates[i];
    __syncthreads();

    const int nT = wave;
    const int nCol = nT * 16 + lm;          // this lane's N column

    for (int it = 0; it < WTA_STEPS; ++it) {
        // C/D layout: VGPR v holds (M = v + 8*half, N = lm) of the 16x16 tile
        v8f c0, c1;
#pragma unroll
        for (int v = 0; v < 8; ++v) {
            c0[v] = xs[( 0 + v + half * 8) * 128 + nCol];   // M-tile 0 (rows 0..15)
            c1[v] = xs[(16 + v + half * 8) * 128 + nCol];   // M-tile 1 (rows 16..31)
        }
        for (int kb = 0; kb < 32; ++kb) {
            const int k0 = kb * 4 + half * 2;
            // B frag (4x16, K x N): W[k][n] = (k==n)?exc:inh — no memory traffic
            v2f b;
            b.x = (k0     == nCol) ? WTA_EXC : WTA_INH;
            b.y = (k0 + 1 == nCol) ? WTA_EXC : WTA_INH;
            // A frags (16x4, M x K): M = lm, K = k0/k0+1
            v2f a0, a1;
            a0.x = xs[lm * 128 + k0];
            a0.y = xs[lm * 128 + k0 + 1];
            a1.x = xs[(16 + lm) * 128 + k0];
            a1.y = xs[(16 + lm) * 128 + k0 + 1];
            c0 = __builtin_amdgcn_wmma_f32_16x16x4_f32(false, a0, false, b,
                                                       (short)0, c0, false, false);
            c1 = __builtin_amdgcn_wmma_f32_16x16x4_f32(false, a1, false, b,
                                                       (short)0, c1, false, false);
        }
#pragma unroll
        for (int v = 0; v < 8; ++v) {
            c0[v] = fminf(fmaxf(c0[v], 0.0f), 1.0f);
            c1[v] = fminf(fmaxf(c1[v], 0.0f), 1.0f);
        }
        __syncthreads();                    // all reads of old x done
#pragma unroll
        for (int v = 0; v < 8; ++v) {
            xs[( 0 + v + half * 8) * 128 + nCol] = c0[v];
            xs[(16 + v + half * 8) * 128 + nCol] = c1[v];
        }
        __syncthreads();                    // writes visible for next iteration
    }
    for (int i = tid; i < 32 * 128; i += 256) rates[i] = xs[i];
}

// ---- Kernel C: J = rates[b,h,t] * V[b,h,t,d], then LIF -> context (262144 elems)
__global__ void av_lif_kernel(const float* __restrict__ rates,
                              const float* __restrict__ V,
                              float* __restrict__ out, int n) {
    int idx = blockIdx.x * blockDim.x + threadIdx.x;
    if (idx >= n) return;
    float J = rates[idx >> 6] * V[idx];     // D = 64
    out[idx] = lif_rate(J);
}

// ---- Kernel D: 4D WTA over n = T*D = 8192 units per (b,h) row.
// W = inh*ones + (exc-inh)*I  =>  x@W^T = inh*S + (exc-inh)*x  (rank-1 + diag),
// turning a 20x 8192x8192 matmul (86 GFLOP, 256 MB of W) into a row reduction.
__global__ __launch_bounds__(256) void wta4d_rank1_kernel(float* __restrict__ x) {
    __shared__ float red[256];
    const int tid = threadIdx.x;
    const size_t base = (size_t)blockIdx.x * 8192;

    float xr[32];
#pragma unroll
    for (int i = 0; i < 32; ++i) xr[i] = x[base + tid + i * 256];

    for (int it = 0; it < WTA_STEPS; ++it) {
        float s = 0.0f;
#pragma unroll
        for (int i = 0; i < 32; ++i) s += xr[i];
        red[tid] = s;
        __syncthreads();
        for (int off = 128; off > 0; off >>= 1) {
            if (tid < off) red[tid] += red[tid + off];
            __syncthreads();
        }
        const float S = red[0];
        __syncthreads();                    // protect red[0] before next write
#pragma unroll
        for (int i = 0; i < 32; ++i) {
            float y = xr[i] + (WTA_INH * S + (WTA_EXC - WTA_INH) * xr[i]);
            xr[i] = fminf(fmaxf(y, 0.0f), 1.0f);
        }
    }
#pragma unroll
    for (int i = 0; i < 32; ++i) x[base + tid + i * 256] = xr[i];
}

extern "C" void kernel_launch(void* const* d_in, const int* in_sizes, int n_in,
                              void* d_out, int out_size, void* d_ws, size_t ws_size,
                              hipStream_t stream) {
    const float* Q = (const float*)d_in[0];
    const float* K = (const float*)d_in[1];
    const float* V = (const float*)d_in[2];
    float* out   = (float*)d_out;           // (B,H,T,D) = 262144 f32
    float* rates = (float*)d_ws;            // 4096 f32 scratch

    qk_lif_kernel<<<16, 256, 0, stream>>>(Q, K, rates, 4096);
    wta3d_wmma_kernel<<<1, 256, 0, stream>>>(rates);
    av_lif_kernel<<<1024, 256, 0, stream>>>(rates, V, out, 262144);
    wta4d_rank1_kernel<<<32, 256, 0, stream>>>(out);
}